// Net_90151363543794
// MI455X (gfx1250) — compile-verified
//
#include <hip/hip_runtime.h>
#include <hip/hip_bf16.h>
#include <math.h>

#define N_NODES    50000
#define N_EDGES    800000
#define NUM_GRAPHS 128
#define TILE_M     32                        // edges per wave (2 WMMA M-tiles)
#define PERM_CAP   (N_EDGES + 16 * TILE_M)   // buckets padded to multiple of 32
#define NUM_TILES  (PERM_CAP / TILE_M)       // 25016 tiles of 32 edges

typedef float v2f __attribute__((ext_vector_type(2)));
typedef float v8f __attribute__((ext_vector_type(8)));

static __device__ __forceinline__ float eluf(float v) {
    return v > 0.f ? v : (expf(v) - 1.f);
}

// ---------------------------------------------------------------------------
// 1) Per-edge spline prep: base bucket (i0 + 4*j0), 4 corner weights,
//    bucket histogram, dst in-degree.
// ---------------------------------------------------------------------------
__global__ void edge_prep_kernel(const float* __restrict__ pseudo,
                                 const int* __restrict__ dst,
                                 float* __restrict__ wq,      // [E,4]
                                 int* __restrict__ bucket,    // [E]
                                 int* __restrict__ cnt16,     // [16]
                                 float* __restrict__ deg)     // [N]
{
    int e = blockIdx.x * blockDim.x + threadIdx.x;
    if (e >= N_EDGES) return;
    float p0 = pseudo[2 * e + 0] * 4.0f;
    float p1 = pseudo[2 * e + 1] * 4.0f;
    int lo0 = (int)floorf(p0); lo0 = lo0 < 0 ? 0 : (lo0 > 3 ? 3 : lo0);
    int lo1 = (int)floorf(p1); lo1 = lo1 < 0 ? 0 : (lo1 > 3 ? 3 : lo1);
    float f0 = p0 - (float)lo0;
    float f1 = p1 - (float)lo1;
    int b = lo0 + 4 * lo1;
    bucket[e] = b;
    // corner order: +0 (w0x*w0y), +5 (w0x*w1y), +1 (w1x*w0y), +6 (w1x*w1y)
    wq[4 * e + 0] = (1.f - f0) * (1.f - f1);
    wq[4 * e + 1] = (1.f - f0) * f1;
    wq[4 * e + 2] = f0 * (1.f - f1);
    wq[4 * e + 3] = f0 * f1;
    atomicAdd(&cnt16[b], 1);
    atomicAdd(&deg[dst[e]], 1.0f);
}

// ---------------------------------------------------------------------------
// 2) Tiny scan over 16 buckets, each padded to a multiple of TILE_M.
// ---------------------------------------------------------------------------
__global__ void bucket_scan_kernel(const int* __restrict__ cnt16,
                                   int* __restrict__ bstart,   // [17]
                                   int* __restrict__ cursor)   // [16]
{
    if (threadIdx.x == 0 && blockIdx.x == 0) {
        int s = 0;
        for (int b = 0; b < 16; ++b) {
            bstart[b] = s;
            cursor[b] = s;
            s += ((cnt16[b] + TILE_M - 1) / TILE_M) * TILE_M;
        }
        bstart[16] = s;
    }
}

// ---------------------------------------------------------------------------
// 3) Scatter edge ids into bucket-sorted order (perm pre-filled with -1).
// ---------------------------------------------------------------------------
__global__ void scatter_perm_kernel(const int* __restrict__ bucket,
                                    int* __restrict__ cursor,
                                    int* __restrict__ perm)
{
    int e = blockIdx.x * blockDim.x + threadIdx.x;
    if (e >= N_EDGES) return;
    int pos = atomicAdd(&cursor[bucket[e]], 1);
    perm[pos] = e;
}

// ---------------------------------------------------------------------------
// 4) Layer-1 edge messages (in_features == 1): pure axpy, no GEMM needed.
// ---------------------------------------------------------------------------
__global__ void layer1_edge_kernel(const float* __restrict__ x,
                                   const int* __restrict__ src,
                                   const int* __restrict__ dst,
                                   const float* __restrict__ wq,
                                   const int* __restrict__ bucket,
                                   const float* __restrict__ W1,   // [25,32]
                                   float* __restrict__ agg)        // [N,32]
{
    int idx = blockIdx.x * blockDim.x + threadIdx.x;
    if (idx >= N_EDGES * 32) return;
    int e  = idx >> 5;
    int ch = idx & 31;
    int b  = bucket[e];
    int kb = (b & 3) + 5 * (b >> 2);
    float w0 = wq[4 * e + 0], w1 = wq[4 * e + 1], w2 = wq[4 * e + 2], w3 = wq[4 * e + 3];
    float acc = w0 * W1[(kb + 0) * 32 + ch]
              + w1 * W1[(kb + 5) * 32 + ch]
              + w2 * W1[(kb + 1) * 32 + ch]
              + w3 * W1[(kb + 6) * 32 + ch];
    float xs = x[src[e]];
    atomicAdd(&agg[(size_t)dst[e] * 32 + ch], xs * acc);
}

// ---------------------------------------------------------------------------
// 5) WMMA edge GEMM for layers 2/3.  One wave handles TILE_M=32 edges
//    (two 16-row M-tiles) sharing one bucket, so every B fragment feeds two
//    V_WMMA_F32_16X16X4_F32 ops:  C[32,64] = sum over 4 corners of
//    (diag(w_c) * Xsrc[32,IC]) @ W[k_c][IC,64].
// ---------------------------------------------------------------------------
template <int IC>
__global__ void __launch_bounds__(32)
edge_gemm_kernel(const float* __restrict__ h,        // [N, IC]
                 const float* __restrict__ Wk,       // [25, IC, 64]
                 const int* __restrict__ perm,       // [PERM_CAP], -1 padded
                 const int* __restrict__ bstart,     // [17]
                 const int* __restrict__ src,
                 const int* __restrict__ dst,
                 const float* __restrict__ wq,       // [E,4]
                 float* __restrict__ agg)            // [N,64]
{
    constexpr int XS = IC + 4;                       // LDS pad: avoid bank conflicts
    __shared__ float Xs[TILE_M * XS];
    __shared__ float wTile[4 * TILE_M];
    __shared__ int   dstTile[TILE_M];
    __shared__ int   sTile[TILE_M];

    const int lane = threadIdx.x;                    // wave32, one wave per block
    const int p0   = blockIdx.x * TILE_M;

    // bucket of this tile (starts are TILE_M-aligned; tiles never straddle)
    int b = 0;
    #pragma unroll
    for (int i = 1; i < 16; ++i) b += (p0 >= bstart[i]) ? 1 : 0;
    const int kb = (b & 3) + 5 * (b >> 2);

    {   // per-row metadata: one row per lane
        int e = perm[p0 + lane];
        int valid = (e >= 0);
        sTile[lane]   = valid ? src[e] : -1;
        dstTile[lane] = valid ? dst[e] : -1;
        float a0 = 0.f, a1 = 0.f, a2 = 0.f, a3 = 0.f;
        if (valid) {
            a0 = wq[4 * e + 0]; a1 = wq[4 * e + 1];
            a2 = wq[4 * e + 2]; a3 = wq[4 * e + 3];
        }
        wTile[0 * TILE_M + lane] = a0;
        wTile[1 * TILE_M + lane] = a1;
        wTile[2 * TILE_M + lane] = a2;
        wTile[3 * TILE_M + lane] = a3;
    }
    __syncthreads();

    // cooperative gather of the 32xIC source-feature tile (float4 per slot)
    constexpr int TOT4 = TILE_M * IC / 4;
    for (int i = lane; i < TOT4; i += 32) {
        int row = i / (IC / 4);
        int c4  = i % (IC / 4);
        int s   = sTile[row];
        float4 v = make_float4(0.f, 0.f, 0.f, 0.f);
        if (s >= 0) v = ((const float4*)(h + (size_t)s * IC))[c4];
        *((float4*)(Xs + row * XS + 4 * c4)) = v;
    }
    __syncthreads();

    const int halfk = lane >> 4;    // K-half selector for 16x4 A / 4x16 B frags
    const int mn    = lane & 15;    // M for A, N for B

    v8f Cacc[2][4];
    #pragma unroll
    for (int t = 0; t < 2; ++t)
        #pragma unroll
        for (int nb = 0; nb < 4; ++nb)
            #pragma unroll
            for (int j = 0; j < 8; ++j) Cacc[t][nb][j] = 0.f;

    const int koff[4] = {0, 5, 1, 6};
    #pragma unroll
    for (int c = 0; c < 4; ++c) {
        const float* Wb = Wk + (size_t)(kb + koff[c]) * IC * 64;
        const float wr0 = wTile[c * TILE_M + mn];         // rows 0..15
        const float wr1 = wTile[c * TILE_M + 16 + mn];    // rows 16..31
        #pragma unroll
        for (int kk = 0; kk < IC / 4; ++kk) {
            const int kq = 4 * kk + 2 * halfk;
            v2f A0, A1;
            A0.x = wr0 * Xs[mn * XS + kq + 0];
            A0.y = wr0 * Xs[mn * XS + kq + 1];
            A1.x = wr1 * Xs[(16 + mn) * XS + kq + 0];
            A1.y = wr1 * Xs[(16 + mn) * XS + kq + 1];
            #pragma unroll
            for (int nb = 0; nb < 4; ++nb) {
                v2f B;                                   // one B frag, two WMMAs
                B.x = Wb[(kq + 0) * 64 + nb * 16 + mn];
                B.y = Wb[(kq + 1) * 64 + nb * 16 + mn];
                Cacc[0][nb] = __builtin_amdgcn_wmma_f32_16x16x4_f32(
                    false, A0, false, B, (short)0, Cacc[0][nb], false, false);
                Cacc[1][nb] = __builtin_amdgcn_wmma_f32_16x16x4_f32(
                    false, A1, false, B, (short)0, Cacc[1][nb], false, false);
            }
        }
    }

    // scatter-add: lane holds C[row = 16*t + j + 8*halfk][col = nb*16 + mn]
    #pragma unroll
    for (int t = 0; t < 2; ++t) {
        #pragma unroll
        for (int nb = 0; nb < 4; ++nb) {
            #pragma unroll
            for (int j = 0; j < 8; ++j) {
                int row = 16 * t + j + 8 * halfk;
                int d = dstTile[row];
                if (d >= 0)
                    atomicAdd(&agg[(size_t)d * 64 + nb * 16 + mn], Cacc[t][nb][j]);
            }
        }
    }
}

// ---------------------------------------------------------------------------
// 6) Node updates: h_out = elu(agg/deg + h_in @ root + bias)
// ---------------------------------------------------------------------------
__global__ void node_update1_kernel(const float* __restrict__ agg,   // [N,32]
                                    const float* __restrict__ deg,
                                    const float* __restrict__ x,     // [N,1]
                                    const float* __restrict__ root,  // [1,32]
                                    const float* __restrict__ bias,  // [32]
                                    float* __restrict__ hout)        // [N,32]
{
    int idx = blockIdx.x * blockDim.x + threadIdx.x;
    if (idx >= N_NODES * 32) return;
    int n = idx >> 5, j = idx & 31;
    float d = deg[n]; d = d < 1.f ? 1.f : d;
    float v = agg[idx] / d + x[n] * root[j] + bias[j];
    hout[idx] = eluf(v);
}

template <int IC>
__global__ void node_update_kernel(const float* __restrict__ agg,    // [N,64]
                                   const float* __restrict__ deg,
                                   const float* __restrict__ hin,    // [N,IC]
                                   const float* __restrict__ root,   // [IC,64]
                                   const float* __restrict__ bias,   // [64]
                                   float* __restrict__ hout)         // [N,64]
{
    int idx = blockIdx.x * blockDim.x + threadIdx.x;
    if (idx >= N_NODES * 64) return;
    int n = idx >> 6, j = idx & 63;
    float d = deg[n]; d = d < 1.f ? 1.f : d;
    float acc = agg[idx] / d + bias[j];
    const float* hr = hin + (size_t)n * IC;
    #pragma unroll 8
    for (int i = 0; i < IC; ++i) acc += hr[i] * root[i * 64 + j];
    hout[idx] = eluf(acc);
}

// ---------------------------------------------------------------------------
// 7) Graph mean-pool
// ---------------------------------------------------------------------------
__global__ void pool_kernel(const float* __restrict__ h,       // [N,64]
                            const int* __restrict__ batch,     // [N]
                            float* __restrict__ pooled,        // [G,64]
                            float* __restrict__ gcnt)          // [G]
{
    int idx = blockIdx.x * blockDim.x + threadIdx.x;
    if (idx >= N_NODES * 64) return;
    int n = idx >> 6, j = idx & 63;
    int g = batch[n];
    atomicAdd(&pooled[(size_t)g * 64 + j], h[idx]);
    if (j == 0) atomicAdd(&gcnt[g], 1.0f);
}

// ---------------------------------------------------------------------------
// 8) FC + log_softmax, one thread per graph
// ---------------------------------------------------------------------------
__global__ void head_kernel(const float* __restrict__ pooled,
                            const float* __restrict__ gcnt,
                            const float* __restrict__ fc_w,    // [64,10]
                            const float* __restrict__ fc_b,    // [10]
                            float* __restrict__ out)           // [G,10]
{
    int g = threadIdx.x;
    if (g >= NUM_GRAPHS) return;
    float c = gcnt[g]; c = c < 1.f ? 1.f : c;
    float feat[64];
    #pragma unroll 8
    for (int i = 0; i < 64; ++i) feat[i] = pooled[(size_t)g * 64 + i] / c;
    float lg[10];
    float m = -1e30f;
    for (int t = 0; t < 10; ++t) {
        float a = fc_b[t];
        #pragma unroll 8
        for (int i = 0; i < 64; ++i) a += feat[i] * fc_w[i * 10 + t];
        lg[t] = a;
        m = a > m ? a : m;
    }
    float se = 0.f;
    for (int t = 0; t < 10; ++t) se += expf(lg[t] - m);
    float lse = m + logf(se);
    for (int t = 0; t < 10; ++t) out[(size_t)g * 10 + t] = lg[t] - lse;
}

// ---------------------------------------------------------------------------
// Launch
// ---------------------------------------------------------------------------
static inline char* carve(char*& p, size_t bytes) {
    char* r = p;
    p += (bytes + 255) & ~(size_t)255;
    return r;
}

extern "C" void kernel_launch(void* const* d_in, const int* in_sizes, int n_in,
                              void* d_out, int out_size, void* d_ws, size_t ws_size,
                              hipStream_t stream)
{
    const float* x      = (const float*)d_in[0];
    const float* pseudo = (const float*)d_in[2];
    const int*   eidx   = (const int*)d_in[3];
    const int*   batch  = (const int*)d_in[4];
    const float* W1     = (const float*)d_in[5];
    const float* root1  = (const float*)d_in[6];
    const float* b1     = (const float*)d_in[7];
    const float* W2     = (const float*)d_in[8];
    const float* root2  = (const float*)d_in[9];
    const float* b2     = (const float*)d_in[10];
    const float* W3     = (const float*)d_in[11];
    const float* root3  = (const float*)d_in[12];
    const float* b3     = (const float*)d_in[13];
    const float* fc_w   = (const float*)d_in[14];
    const float* fc_b   = (const float*)d_in[15];
    float* out = (float*)d_out;

    const int* src = eidx;
    const int* dst = eidx + N_EDGES;

    char* p = (char*)d_ws;
    float* wq      = (float*)carve(p, (size_t)N_EDGES * 4 * sizeof(float));
    int*   bucket  = (int*)  carve(p, (size_t)N_EDGES * sizeof(int));
    int*   perm    = (int*)  carve(p, (size_t)PERM_CAP * sizeof(int));
    int*   cnt16   = (int*)  carve(p, 16 * sizeof(int));
    int*   bstart  = (int*)  carve(p, 17 * sizeof(int));
    int*   cursor  = (int*)  carve(p, 16 * sizeof(int));
    float* deg     = (float*)carve(p, (size_t)N_NODES * sizeof(float));
    float* agg     = (float*)carve(p, (size_t)N_NODES * 64 * sizeof(float));
    float* h1      = (float*)carve(p, (size_t)N_NODES * 32 * sizeof(float));
    float* h2      = (float*)carve(p, (size_t)N_NODES * 64 * sizeof(float));
    float* h3      = (float*)carve(p, (size_t)N_NODES * 64 * sizeof(float));
    float* pooled  = (float*)carve(p, (size_t)NUM_GRAPHS * 64 * sizeof(float));
    float* gcnt    = (float*)carve(p, (size_t)NUM_GRAPHS * sizeof(float));

    // --- spline prep + bucketing ---
    hipMemsetAsync(cnt16, 0, 16 * sizeof(int), stream);
    hipMemsetAsync(deg, 0, (size_t)N_NODES * sizeof(float), stream);
    edge_prep_kernel<<<(N_EDGES + 255) / 256, 256, 0, stream>>>(
        pseudo, dst, wq, bucket, cnt16, deg);
    bucket_scan_kernel<<<1, 32, 0, stream>>>(cnt16, bstart, cursor);
    hipMemsetAsync(perm, 0xFF, (size_t)PERM_CAP * sizeof(int), stream);  // -1 fill
    scatter_perm_kernel<<<(N_EDGES + 255) / 256, 256, 0, stream>>>(bucket, cursor, perm);

    // --- layer 1 (in=1 -> 32) ---
    hipMemsetAsync(agg, 0, (size_t)N_NODES * 32 * sizeof(float), stream);
    layer1_edge_kernel<<<((size_t)N_EDGES * 32 + 255) / 256, 256, 0, stream>>>(
        x, src, dst, wq, bucket, W1, agg);
    node_update1_kernel<<<(N_NODES * 32 + 255) / 256, 256, 0, stream>>>(
        agg, deg, x, root1, b1, h1);

    // --- layer 2 (32 -> 64), WMMA ---
    hipMemsetAsync(agg, 0, (size_t)N_NODES * 64 * sizeof(float), stream);
    edge_gemm_kernel<32><<<NUM_TILES, 32, 0, stream>>>(
        h1, W2, perm, bstart, src, dst, wq, agg);
    node_update_kernel<32><<<(N_NODES * 64 + 255) / 256, 256, 0, stream>>>(
        agg, deg, h1, root2, b2, h2);

    // --- layer 3 (64 -> 64), WMMA ---
    hipMemsetAsync(agg, 0, (size_t)N_NODES * 64 * sizeof(float), stream);
    edge_gemm_kernel<64><<<NUM_TILES, 32, 0, stream>>>(
        h2, W3, perm, bstart, src, dst, wq, agg);
    node_update_kernel<64><<<(N_NODES * 64 + 255) / 256, 256, 0, stream>>>(
        agg, deg, h2, root3, b3, h3);

    // --- pool + head ---
    hipMemsetAsync(pooled, 0, (size_t)NUM_GRAPHS * 64 * sizeof(float), stream);
    hipMemsetAsync(gcnt, 0, (size_t)NUM_GRAPHS * sizeof(float), stream);
    pool_kernel<<<(N_NODES * 64 + 255) / 256, 256, 0, stream>>>(h3, batch, pooled, gcnt);
    head_kernel<<<1, 128, 0, stream>>>(pooled, gcnt, fc_w, fc_b, out);

    (void)in_sizes; (void)n_in; (void)out_size; (void)ws_size;
}